// CCX_loss_1580547971557
// MI455X (gfx1250) — compile-verified
//
#include <hip/hip_runtime.h>
#include <hip/hip_bf16.h>
#include <math.h>

typedef __attribute__((ext_vector_type(2))) float v2f;
typedef __attribute__((ext_vector_type(8))) float v8f;

#define CCH 256      // channels (K dim)
#define HWD 4096     // H*W (spatial, GEMM M/N dim)
#define NBT 8        // batch
#define CEPS 1e-6f
#define INV_H 2.0f   // 1 / h_bw
#define FLT_BIG 3.402823e38f

__device__ __forceinline__ v8f wmma4(v2f a, v2f b, v8f c) {
    return __builtin_amdgcn_wmma_f32_16x16x4_f32(false, a, false, b, (short)0, c, false, false);
}

// ---------------------------------------------------------------------------
// Layouts:
//   Xn : (N, HW, C) row-major                      -> A fragments: contiguous k-pairs
//   Yp : (N, C/2, HW, 2) "k-pair interleaved" Y^T  -> B fragments: contiguous k-pairs
// WMMA f32 16x16x4 (ISA 7.12.2): lanes 0-15 hold K=k0,k0+1; lanes 16-31 K=k0+2,k0+3.
// C/D: VGPR r -> row r + 8*(lane>=16), col lane&15.
// ---------------------------------------------------------------------------

// Stage 0: per-channel mean of y over (N,H,W)
__global__ void ccx_ymu_kernel(const float* __restrict__ y, float* __restrict__ ymu) {
    __shared__ float s[256];
    const int c = blockIdx.x;
    float acc = 0.f;
    for (int n = 0; n < NBT; ++n) {
        const float* b = y + ((size_t)n * CCH + c) * HWD;
        for (int t = threadIdx.x; t < HWD; t += 256) acc += b[t];
    }
    s[threadIdx.x] = acc;
    __syncthreads();
    for (int off = 128; off > 0; off >>= 1) {
        if (threadIdx.x < off) s[threadIdx.x] += s[threadIdx.x + off];
        __syncthreads();
    }
    if (threadIdx.x == 0) ymu[c] = s[0] / (float)(NBT * HWD);
}

// Stage 1: center by ymu, L2-normalize over channels; emit Xn and Yp.
__global__ void ccx_normalize_kernel(const float* __restrict__ x, const float* __restrict__ y,
                                     const float* __restrict__ ymu,
                                     float* __restrict__ Xn, float* __restrict__ Yp) {
    const size_t p  = (size_t)blockIdx.x * 256 + threadIdx.x;   // 0 .. N*HW-1
    const int   n   = (int)(p >> 12);
    const int   hw  = (int)(p & (HWD - 1));
    const float* xb = x + (size_t)n * CCH * HWD + hw;
    const float* yb = y + (size_t)n * CCH * HWD + hw;
    float sx = 0.f, sy = 0.f;
    for (int c = 0; c < CCH; ++c) {
        const float mu = ymu[c];
        const float xv = xb[(size_t)c * HWD] - mu;
        const float yv = yb[(size_t)c * HWD] - mu;
        sx += xv * xv;
        sy += yv * yv;
    }
    const float rx = 1.0f / sqrtf(sx);
    const float ry = 1.0f / sqrtf(sy);
    float* xo = Xn + p * CCH;
    float* yo = Yp + (size_t)n * CCH * HWD;
    for (int c = 0; c < CCH; ++c) {
        const float mu = ymu[c];
        xo[c] = (xb[(size_t)c * HWD] - mu) * rx;
        yo[((size_t)(c >> 1) * HWD + hw) * 2 + (c & 1)] = (yb[(size_t)c * HWD] - mu) * ry;
    }
}

// ---------------------------------------------------------------------------
// Stage 2: d_min[i] = min_j (1 - cos(i,j)).  WG owns 32 rows; wave computes
// two stacked 16x16 tiles per j-tile sharing one B fragment.
// ---------------------------------------------------------------------------
__global__ void __launch_bounds__(256)
ccx_min_kernel(const float* __restrict__ Xn, const float* __restrict__ Yp,
               float* __restrict__ dmin) {
    __shared__ float Alds[32 * CCH];          // 32 KB
    __shared__ float red[8 * 32 * 16];        // 16 KB: [wave][lane][tile*8+r]
    const int n  = blockIdx.x >> 7;
    const int i0 = (blockIdx.x & 127) << 5;
    const float* Xb = Xn + ((size_t)n * HWD + i0) * CCH;
    const float* Yb = Yp + (size_t)n * CCH * HWD;
    for (int t = threadIdx.x; t < 32 * CCH; t += 256) Alds[t] = Xb[t];
    __syncthreads();

    const int lane = threadIdx.x & 31;
    const int wv   = threadIdx.x >> 5;
    const int m    = lane & 15;
    const int hi   = lane >> 4;
    float rmin[16];
    #pragma unroll
    for (int r = 0; r < 16; ++r) rmin[r] = FLT_BIG;

    for (int j0 = wv * 16; j0 < HWD; j0 += 128) {
        v8f c0 = {}, c1 = {};
        #pragma unroll 4
        for (int k0 = 0; k0 < CCH; k0 += 4) {
            const int kp = (k0 >> 1) + hi;               // k-pair index
            const v2f b  = *(const v2f*)(Yb + ((size_t)kp * HWD + j0 + m) * 2);
            const v2f a0 = *(const v2f*)(Alds + m * CCH + k0 + 2 * hi);
            const v2f a1 = *(const v2f*)(Alds + (16 + m) * CCH + k0 + 2 * hi);
            c0 = wmma4(a0, b, c0);
            c1 = wmma4(a1, b, c1);
        }
        #pragma unroll
        for (int r = 0; r < 8; ++r) {
            rmin[r]     = fminf(rmin[r],     1.0f - c0[r]);
            rmin[8 + r] = fminf(rmin[8 + r], 1.0f - c1[r]);
        }
    }
    #pragma unroll
    for (int r = 0; r < 16; ++r) red[(wv * 32 + lane) * 16 + r] = rmin[r];
    __syncthreads();

    if (threadIdx.x < 32) {                   // fixed-order -> deterministic
        const int row = threadIdx.x;          // 0..31 within block
        const int t  = row >> 4;
        const int mm = row & 15;
        const int r  = mm & 7;
        const int h  = mm >> 3;
        float mn = FLT_BIG;
        for (int w2 = 0; w2 < 8; ++w2)
            for (int l = 0; l < 16; ++l)
                mn = fminf(mn, red[((w2 * 32) + (h << 4) + l) * 16 + t * 8 + r]);
        dmin[(size_t)n * HWD + i0 + row] = mn;
    }
}

// ---------------------------------------------------------------------------
// Stage 3: rowsum[i] = sum_j exp(2*(1 - d/(dmin+eps)))
// ---------------------------------------------------------------------------
__global__ void __launch_bounds__(256)
ccx_sum_kernel(const float* __restrict__ Xn, const float* __restrict__ Yp,
               const float* __restrict__ dmin, float* __restrict__ rowsum) {
    __shared__ float Alds[32 * CCH];
    __shared__ float red[8 * 32 * 16];
    __shared__ float dml[32];
    const int n  = blockIdx.x >> 7;
    const int i0 = (blockIdx.x & 127) << 5;
    const float* Xb = Xn + ((size_t)n * HWD + i0) * CCH;
    const float* Yb = Yp + (size_t)n * CCH * HWD;
    for (int t = threadIdx.x; t < 32 * CCH; t += 256) Alds[t] = Xb[t];
    if (threadIdx.x < 32) dml[threadIdx.x] = dmin[(size_t)n * HWD + i0 + threadIdx.x] + CEPS;
    __syncthreads();

    const int lane = threadIdx.x & 31;
    const int wv   = threadIdx.x >> 5;
    const int m    = lane & 15;
    const int hi   = lane >> 4;
    float dmv[16], rs[16];
    #pragma unroll
    for (int r = 0; r < 8; ++r) {
        dmv[r]     = dml[r + 8 * hi];
        dmv[8 + r] = dml[16 + r + 8 * hi];
        rs[r] = 0.f; rs[8 + r] = 0.f;
    }

    for (int j0 = wv * 16; j0 < HWD; j0 += 128) {
        v8f c0 = {}, c1 = {};
        #pragma unroll 4
        for (int k0 = 0; k0 < CCH; k0 += 4) {
            const int kp = (k0 >> 1) + hi;
            const v2f b  = *(const v2f*)(Yb + ((size_t)kp * HWD + j0 + m) * 2);
            const v2f a0 = *(const v2f*)(Alds + m * CCH + k0 + 2 * hi);
            const v2f a1 = *(const v2f*)(Alds + (16 + m) * CCH + k0 + 2 * hi);
            c0 = wmma4(a0, b, c0);
            c1 = wmma4(a1, b, c1);
        }
        #pragma unroll
        for (int r = 0; r < 8; ++r) {
            rs[r]     += expf(INV_H * (1.0f - (1.0f - c0[r]) / dmv[r]));
            rs[8 + r] += expf(INV_H * (1.0f - (1.0f - c1[r]) / dmv[8 + r]));
        }
    }
    #pragma unroll
    for (int r = 0; r < 16; ++r) red[(wv * 32 + lane) * 16 + r] = rs[r];
    __syncthreads();

    if (threadIdx.x < 32) {                   // fixed-order -> deterministic
        const int row = threadIdx.x;
        const int t  = row >> 4;
        const int mm = row & 15;
        const int r  = mm & 7;
        const int h  = mm >> 3;
        float s = 0.f;
        for (int w2 = 0; w2 < 8; ++w2)
            for (int l = 0; l < 16; ++l)
                s += red[((w2 * 32) + (h << 4) + l) * 16 + t * 8 + r];
        rowsum[(size_t)n * HWD + i0 + row] = s;
    }
}

// ---------------------------------------------------------------------------
// Stage 4: WG owns 32 columns; wave computes two side-by-side 16x16 tiles per
// i-tile sharing one A fragment.  M[j] = max_i w(i,j)/rowsum[i]; emit per-WG
// partial sum of the 32 column maxima.
// ---------------------------------------------------------------------------
__global__ void __launch_bounds__(256)
ccx_colmax_kernel(const float* __restrict__ Xn, const float* __restrict__ Yp,
                  const float* __restrict__ dmin, const float* __restrict__ rowsum,
                  float* __restrict__ colpart) {
    __shared__ float Blds[CCH * 32];          // 32 KB: [(k/2)][q][2]
    __shared__ float red[8 * 32 * 2];
    __shared__ float cm[32];
    const int n   = blockIdx.x >> 7;
    const int jbk = blockIdx.x & 127;
    const int j0  = jbk << 5;
    const float* Xg = Xn + (size_t)n * HWD * CCH;
    const float* Yb = Yp + (size_t)n * CCH * HWD;
    // stage B block: Blds[(kk*32 + q)*2 + par] = Yp[(kk*HW + j0 + q)*2 + par]
    for (int t = threadIdx.x; t < CCH * 32; t += 256) {
        const int kk  = t >> 6;               // k-pair 0..127
        const int rem = t & 63;               // q*2 + par
        Blds[kk * 64 + rem] = Yb[(size_t)kk * (HWD * 2) + j0 * 2 + rem];
    }
    __syncthreads();

    const int lane = threadIdx.x & 31;
    const int wv   = threadIdx.x >> 5;
    const int m    = lane & 15;
    const int hi   = lane >> 4;
    float cmax0 = 0.f, cmax1 = 0.f;           // ccx values strictly positive

    for (int i0 = wv * 16; i0 < HWD; i0 += 128) {
        v8f c0 = {}, c1 = {};
        #pragma unroll 4
        for (int k0 = 0; k0 < CCH; k0 += 4) {
            const int kp = (k0 >> 1) + hi;
            const v2f a  = *(const v2f*)(Xg + (size_t)(i0 + m) * CCH + k0 + 2 * hi);
            const v2f b0 = *(const v2f*)(Blds + (kp * 32 + m) * 2);
            const v2f b1 = *(const v2f*)(Blds + (kp * 32 + 16 + m) * 2);
            c0 = wmma4(a, b0, c0);
            c1 = wmma4(a, b1, c1);
        }
        #pragma unroll
        for (int r = 0; r < 8; ++r) {
            const size_t i = (size_t)n * HWD + i0 + r + 8 * hi;
            const float idm = 1.0f / (dmin[i] + CEPS);
            const float irs = 1.0f / rowsum[i];
            cmax0 = fmaxf(cmax0, expf(INV_H * (1.0f - (1.0f - c0[r]) * idm)) * irs);
            cmax1 = fmaxf(cmax1, expf(INV_H * (1.0f - (1.0f - c1[r]) * idm)) * irs);
        }
    }
    red[(wv * 32 + lane) * 2 + 0] = cmax0;
    red[(wv * 32 + lane) * 2 + 1] = cmax1;
    __syncthreads();

    if (threadIdx.x < 32) {
        const int q  = threadIdx.x;           // column within block 0..31
        const int t  = q >> 4;
        const int qq = q & 15;
        float mx = 0.f;
        for (int w2 = 0; w2 < 8; ++w2) {
            mx = fmaxf(mx, red[((w2 * 32) + qq) * 2 + t]);
            mx = fmaxf(mx, red[((w2 * 32) + 16 + qq) * 2 + t]);
        }
        cm[q] = mx;
    }
    __syncthreads();
    if (threadIdx.x == 0) {
        float s = 0.f;
        for (int q = 0; q < 32; ++q) s += cm[q];
        colpart[(size_t)n * 128 + jbk] = s;
    }
}

// Stage 5: loss = mean_n( -log( (sum_j M[j])/HW + eps ) )
__global__ void ccx_final_kernel(const float* __restrict__ colpart, float* __restrict__ out) {
    __shared__ float s[NBT];
    if (threadIdx.x < NBT) {
        float acc = 0.f;
        for (int j = 0; j < 128; ++j) acc += colpart[(size_t)threadIdx.x * 128 + j];
        s[threadIdx.x] = -logf(acc / (float)HWD + CEPS);
    }
    __syncthreads();
    if (threadIdx.x == 0) {
        float l = 0.f;
        for (int n = 0; n < NBT; ++n) l += s[n];
        out[0] = l / (float)NBT;
    }
}

// ---------------------------------------------------------------------------
extern "C" void kernel_launch(void* const* d_in, const int* in_sizes, int n_in,
                              void* d_out, int out_size, void* d_ws, size_t ws_size,
                              hipStream_t stream) {
    (void)in_sizes; (void)n_in; (void)out_size; (void)ws_size;
    const float* x = (const float*)d_in[0];
    const float* y = (const float*)d_in[1];

    float* ws      = (float*)d_ws;
    const size_t XN_ELTS = (size_t)NBT * HWD * CCH;    // 8,388,608
    float* Xn      = ws;
    float* Yp      = Xn + XN_ELTS;
    float* ymu     = Yp + XN_ELTS;
    float* dmin    = ymu + CCH;
    float* rowsum  = dmin + (size_t)NBT * HWD;
    float* colpart = rowsum + (size_t)NBT * HWD;       // N*128 partials

    ccx_ymu_kernel      <<<CCH,               256, 0, stream>>>(y, ymu);
    ccx_normalize_kernel<<<(NBT * HWD) / 256, 256, 0, stream>>>(x, y, ymu, Xn, Yp);
    ccx_min_kernel      <<<NBT * (HWD / 32),  256, 0, stream>>>(Xn, Yp, dmin);
    ccx_sum_kernel      <<<NBT * (HWD / 32),  256, 0, stream>>>(Xn, Yp, dmin, rowsum);
    ccx_colmax_kernel   <<<NBT * (HWD / 32),  256, 0, stream>>>(Xn, Yp, dmin, rowsum, colpart);
    ccx_final_kernel    <<<1,                 256, 0, stream>>>(colpart, (float*)d_out);
}